// DVNDTA_5755256177241
// MI455X (gfx1250) — compile-verified
//
#include <hip/hip_runtime.h>
#include <hip/hip_bf16.h>

// ---------------- problem constants (match reference) ----------------
#define NN       100000
#define E_INTRA  800000
#define E_INTER  400000
#define NODE_DIM 35
#define HD       128
#define GG       256
#define NLAYERS  4

typedef float v2f __attribute__((ext_vector_type(2)));
typedef float v8f __attribute__((ext_vector_type(8)));

__device__ __forceinline__ float silu_f(float x) {
    return x / (1.0f + expf(-x));
}

// ---------------- zero fill ----------------
__global__ void zero_kernel(float* __restrict__ p, size_t count) {
    size_t stride = (size_t)gridDim.x * blockDim.x;
    for (size_t i = (size_t)blockIdx.x * blockDim.x + threadIdx.x; i < count; i += stride)
        p[i] = 0.0f;
}

// ---------------- input layer: h = silu(x @ Wn + bn), K = 35 ----------------
__global__ __launch_bounds__(128)
void lin_node_kernel(const float* __restrict__ x, const float* __restrict__ Wn,
                     const float* __restrict__ bn, float* __restrict__ h) {
    __shared__ float xs[NODE_DIM];
    int n = blockIdx.x;
    int t = threadIdx.x;
    if (t < NODE_DIM) xs[t] = x[(size_t)n * NODE_DIM + t];
    __syncthreads();
    float acc = bn[t];
#pragma unroll
    for (int k = 0; k < NODE_DIM; ++k)
        acc = fmaf(xs[k], Wn[k * HD + t], acc);
    h[(size_t)n * HD + t] = silu_f(acc);
}

// ---------------- degree count (scatter-add ones) ----------------
__global__ void count_kernel(const int* __restrict__ dst, int E, float* __restrict__ deg) {
    int i = blockIdx.x * blockDim.x + threadIdx.x;
    if (i < E) atomicAdd(&deg[dst[i]], 1.0f);
}

// ---------------- per-node epilogue scales ----------------
// intra: s = 1/(deg+1), bias-mul = deg.  inter: s = log(cnt+1), bias-mul = cnt.
__global__ void node_scale_kernel(const float* __restrict__ bmI, const float* __restrict__ bmE,
                                  float* __restrict__ sI, float* __restrict__ sE) {
    int i = blockIdx.x * blockDim.x + threadIdx.x;
    if (i < NN) {
        sI[i] = 1.0f / (bmI[i] + 1.0f);
        sE[i] = logf(bmE[i] + 1.0f);
    }
}

// ---------------- inter-edge geometric weight: exp(-||p_s - p_d||^2) ----------------
__global__ void winter_kernel(const int* __restrict__ ei, const float* __restrict__ pos,
                              float* __restrict__ w) {
    int i = blockIdx.x * blockDim.x + threadIdx.x;
    if (i >= E_INTER) return;
    int s = ei[i];
    int d = ei[E_INTER + i];
    float dx = pos[3 * s + 0] - pos[3 * d + 0];
    float dy = pos[3 * s + 1] - pos[3 * d + 1];
    float dz = pos[3 * s + 2] - pos[3 * d + 2];
    w[i] = expf(-(dx * dx + dy * dy + dz * dz));
}

// ---------------- edge scatter: agg[dst] += h[src] * wgt[e]  (wave per edge) ----------------
__global__ __launch_bounds__(256)
void scatter_kernel(const float* __restrict__ h, const int* __restrict__ src,
                    const int* __restrict__ dst, const float* __restrict__ wgt,
                    float* __restrict__ agg, int E) {
    int gw   = (blockIdx.x * 256 + threadIdx.x) >> 5;  // global wave id = edge
    int lane = threadIdx.x & 31;
    if (gw >= E) return;
    int s = src[gw];
    int d = dst[gw];
    float w = wgt[gw];
    const float4 v = *reinterpret_cast<const float4*>(h + (size_t)s * HD + lane * 4);
    float* base = agg + (size_t)d * HD + lane * 4;
    atomicAdd(base + 0, v.x * w);
    atomicAdd(base + 1, v.y * w);
    atomicAdd(base + 2, v.z * w);
    atomicAdd(base + 3, v.w * w);
}

// ---------------- WMMA GEMM: val = (A @ W + bm[r]*bias[c]) * sc[r] ----------------
// mode 0: out[r,c] = val                       (produces m_i)
// mode 1: vp = silu(mi+vp); vl = silu(val+vl); h += vp+vl   (fused layer update)
//
// 256 threads = 8 waves; block covers 64 rows x 128 cols; each wave owns one
// 16-col tile and loops 4 row tiles.  K = 128 = 32 steps of V_WMMA_F32_16X16X4_F32.
#define LDA 132   // LDS row stride in floats; 132 % 64 == 4 -> conflict-free column reads
__global__ __launch_bounds__(256)
void gemm_node_kernel(const float* __restrict__ A, const float* __restrict__ W,
                      const float* __restrict__ bias, const float* __restrict__ bm,
                      const float* __restrict__ sc, float* __restrict__ out,
                      const float* __restrict__ mi, float* __restrict__ vp,
                      float* __restrict__ vl, float* __restrict__ h,
                      int nrows, int mode) {
    __shared__ float As[64 * LDA];

    const int tid  = threadIdx.x;
    const int wave = tid >> 5;
    const int lane = tid & 31;
    const int r0   = blockIdx.x * 64;

    // cooperative load of 64x128 A tile into LDS (zero-pad past nrows)
#pragma unroll
    for (int it = 0; it < 8; ++it) {
        int idx = tid + it * 256;       // 0..2047
        int row = idx >> 5;             // 0..63
        int c4  = (idx & 31) * 4;       // 0..124
        int r   = r0 + row;
        float4 v = make_float4(0.f, 0.f, 0.f, 0.f);
        if (r < nrows) v = *reinterpret_cast<const float4*>(A + (size_t)r * HD + c4);
        *reinterpret_cast<float4*>(&As[row * LDA + c4]) = v;
    }
    __syncthreads();

    const int n0    = wave * 16;        // column tile base
    const int kl    = lane & 15;        // row (A frag) / col (B frag) within tile
    const int khalf = (lane >> 4) * 2;  // lanes 0-15: K = {k,k+1}; lanes 16-31: K = {k+2,k+3}

    // preload all 32 B fragments for this column tile (W is L1/L2 resident)
    v2f breg[32];
#pragma unroll
    for (int s = 0; s < 32; ++s) {
        int k = 4 * s + khalf;
        v2f b;
        b.x = W[(k + 0) * HD + n0 + kl];
        b.y = W[(k + 1) * HD + n0 + kl];
        breg[s] = b;
    }

    const int   mbase = (lane >> 4) * 8;  // C/D layout: lanes 0-15 -> M=0..7, 16-31 -> M=8..15
    const int   col   = n0 + kl;
    const float bcol  = bias[col];

    for (int rt = 0; rt < 4; ++rt) {
        v8f c = {0.f, 0.f, 0.f, 0.f, 0.f, 0.f, 0.f, 0.f};
        const float* arow = &As[(rt * 16 + kl) * LDA];
#pragma unroll
        for (int s = 0; s < 32; ++s) {
            int k = 4 * s + khalf;
            v2f a;
            a.x = arow[k];
            a.y = arow[k + 1];
            c = __builtin_amdgcn_wmma_f32_16x16x4_f32(
                    false, a, false, breg[s], (short)0, c, false, false);
        }
#pragma unroll
        for (int j = 0; j < 8; ++j) {
            int r = r0 + rt * 16 + mbase + j;
            if (r < nrows) {
                size_t idx = (size_t)r * HD + col;
                float val = (c[j] + bm[r] * bcol) * sc[r];
                if (mode == 0) {
                    out[idx] = val;
                } else {
                    float nvp = silu_f(mi[idx] + vp[idx]);
                    float nvl = silu_f(val + vl[idx]);
                    vp[idx] = nvp;
                    vl[idx] = nvl;
                    h[idx] += nvp + nvl;
                }
            }
        }
    }
}

// ---------------- global add pool over sorted batch ----------------
__global__ __launch_bounds__(128)
void pool_kernel(const float* __restrict__ h, const int* __restrict__ batch,
                 float* __restrict__ g) {
    int n = blockIdx.x;
    int t = threadIdx.x;
    atomicAdd(&g[(size_t)batch[n] * HD + t], h[(size_t)n * HD + t]);
}

// ---------------- FC head: 3x (linear + leaky_relu + BN affine) + output linear ----------------
__global__ __launch_bounds__(128)
void fc_head_kernel(const float* __restrict__ g, const float* __restrict__ fcW,
                    const float* __restrict__ fcb, const float* __restrict__ gamma,
                    const float* __restrict__ beta, const float* __restrict__ outW,
                    const float* __restrict__ outb, float* __restrict__ out) {
    __shared__ float cur[HD];
    __shared__ float red[HD];
    const int gi = blockIdx.x;
    const int t  = threadIdx.x;
    const float bnsc = rsqrtf(1.0f + 1e-5f);

    cur[t] = g[(size_t)gi * HD + t];
    __syncthreads();
    for (int j = 0; j < 3; ++j) {
        const float* Wj = fcW + (size_t)j * HD * HD;
        float acc = fcb[j * HD + t];
        for (int k = 0; k < HD; ++k)
            acc = fmaf(cur[k], Wj[k * HD + t], acc);
        acc = (acc > 0.0f) ? acc : 0.01f * acc;          // leaky relu
        acc = acc * bnsc * gamma[j * HD + t] + beta[j * HD + t];
        __syncthreads();
        cur[t] = acc;
        __syncthreads();
    }
    red[t] = cur[t] * outW[t];
    __syncthreads();
    for (int s = 64; s > 0; s >>= 1) {
        if (t < s) red[t] += red[t + s];
        __syncthreads();
    }
    if (t == 0) out[gi] = red[0] + outb[0];
}

// ---------------- host orchestration ----------------
extern "C" void kernel_launch(void* const* d_in, const int* in_sizes, int n_in,
                              void* d_out, int out_size, void* d_ws, size_t ws_size,
                              hipStream_t stream) {
    (void)in_sizes; (void)n_in; (void)out_size; (void)ws_size;

    const float* x        = (const float*)d_in[0];
    const int*   eiI      = (const int*)  d_in[1];   // [2, E_INTRA]: src row then dst row
    const int*   eiE      = (const int*)  d_in[2];   // [2, E_INTER]
    const float* pos      = (const float*)d_in[3];
    const float* eattr    = (const float*)d_in[4];   // [E_INTRA]
    const int*   batch    = (const int*)  d_in[5];
    const float* lnW      = (const float*)d_in[6];
    const float* lnb      = (const float*)d_in[7];
    const float* W_intra  = (const float*)d_in[8];   // [4,128,128]
    const float* b_intra  = (const float*)d_in[9];
    const float* W_inter  = (const float*)d_in[10];
    const float* b_inter  = (const float*)d_in[11];
    const float* fcW      = (const float*)d_in[12];
    const float* fcb      = (const float*)d_in[13];
    const float* bngamma  = (const float*)d_in[14];
    const float* bnbeta   = (const float*)d_in[15];
    const float* outW     = (const float*)d_in[16];
    const float* outb     = (const float*)d_in[17];
    float*       out      = (float*)d_out;

    const size_t NH = (size_t)NN * HD;
    float* ws   = (float*)d_ws;
    float* h    = ws;
    float* vp   = ws + 1 * NH;
    float* vl   = ws + 2 * NH;   // vp,vl contiguous
    float* aggI = ws + 3 * NH;
    float* aggE = ws + 4 * NH;   // aggI,aggE contiguous
    float* mi   = ws + 5 * NH;
    float* bmI  = ws + 6 * NH;
    float* bmE  = bmI + NN;      // bmI,bmE contiguous
    float* sI   = bmI + 2 * (size_t)NN;
    float* sE   = bmI + 3 * (size_t)NN;
    float* wint = bmI + 4 * (size_t)NN;
    float* gp   = wint + E_INTER;

    const int ZB = 2048;

    // init state
    zero_kernel<<<ZB, 256, 0, stream>>>(vp, 2 * NH);              // v_p, v_l = 0
    zero_kernel<<<ZB, 256, 0, stream>>>(bmI, 2 * (size_t)NN);     // degree counters = 0
    zero_kernel<<<ZB, 256, 0, stream>>>(gp, (size_t)GG * HD);     // pool accum = 0

    // h = silu(x @ lnW + lnb)
    lin_node_kernel<<<NN, 128, 0, stream>>>(x, lnW, lnb, h);

    // degrees + per-node epilogue scales + inter edge weights
    count_kernel<<<(E_INTRA + 255) / 256, 256, 0, stream>>>(eiI + E_INTRA, E_INTRA, bmI);
    count_kernel<<<(E_INTER + 255) / 256, 256, 0, stream>>>(eiE + E_INTER, E_INTER, bmE);
    node_scale_kernel<<<(NN + 255) / 256, 256, 0, stream>>>(bmI, bmE, sI, sE);
    winter_kernel<<<(E_INTER + 255) / 256, 256, 0, stream>>>(eiE, pos, wint);

    const int gemm_blocks = (NN + 63) / 64;
    for (int l = 0; l < NLAYERS; ++l) {
        zero_kernel<<<ZB, 256, 0, stream>>>(aggI, 2 * NH);  // aggI + aggE

        scatter_kernel<<<(E_INTRA + 7) / 8, 256, 0, stream>>>(
            h, eiI, eiI + E_INTRA, eattr, aggI, E_INTRA);
        scatter_kernel<<<(E_INTER + 7) / 8, 256, 0, stream>>>(
            h, eiE, eiE + E_INTER, wint, aggE, E_INTER);

        // m_i = (aggI @ W_intra + deg*b) / (deg+1)
        gemm_node_kernel<<<gemm_blocks, 256, 0, stream>>>(
            aggI, W_intra + (size_t)l * HD * HD, b_intra + l * HD,
            bmI, sI, mi, mi, vp, vl, h, NN, 0);
        // m_e = (aggE @ W_inter + cnt*b) * log(cnt+1); fused: vp/vl/h update
        gemm_node_kernel<<<gemm_blocks, 256, 0, stream>>>(
            aggE, W_inter + (size_t)l * HD * HD, b_inter + l * HD,
            bmE, sE, mi, mi, vp, vl, h, NN, 1);
    }

    // pool + FC head
    pool_kernel<<<NN, 128, 0, stream>>>(h, batch, gp);
    fc_head_kernel<<<GG, 128, 0, stream>>>(gp, fcW, fcb, bngamma, bnbeta, outW, outb, out);
}